// CrossEntropyLoss_54640573940344
// MI455X (gfx1250) — compile-verified
//
#include <hip/hip_runtime.h>
#include <stdint.h>

// Reference collapses to one scalar:
//   out = exp(x[0, label[0]]) / sum_j exp(x[0, j])
// Only row 0 (32000 fp32 = 128 KB) is live -> single-workgroup,
// launch-latency-bound kernel. Data movement uses the CDNA5 async
// global->LDS path (ASYNCcnt); compute is fp32 VALU + v_exp_f32.
// No matmul content, so no WMMA by design.

#define N_COLS 32000
#define VEC4   (N_COLS / 4)        // 8000 float4 transfers
#define BLOCK  1024                // 32 wave32s on one WGP
#define WAVES  (BLOCK / 32)
#define FULL_ITERS (VEC4 / BLOCK)  // 7 unconditional iterations
#define TAIL_BASE  (FULL_ITERS * BLOCK)          // 7168
#define TAIL_COUNT (VEC4 - TAIL_BASE)            // 832 threads do an 8th

__global__ __launch_bounds__(BLOCK, 1)
void softmax_row0_prob_kernel(const float* __restrict__ x,
                              const int*  __restrict__ label,
                              float* __restrict__ out)
{
    // Static LDS: compile-time check that 128,128 B/workgroup is legal on gfx1250.
    __shared__ float lds_row[N_COLS];
    __shared__ float wsum[WAVES];

    const int tid = threadIdx.x;

    const float4* gsrc = reinterpret_cast<const float4*>(x);
    float4*       ldst = reinterpret_cast<float4*>(lds_row);

    // ---- Stage row 0 (128 KB) into LDS: 7 unconditional + 1 guarded async
    // b128 issues per thread, back-to-back, no destination VGPRs consumed.
    #pragma unroll
    for (int k = 0; k < FULL_ITERS; ++k) {
        const int idx = tid + k * BLOCK;
        uint64_t ga = (uint64_t)(uintptr_t)(gsrc + idx);
        uint32_t la = (uint32_t)(uintptr_t)(ldst + idx);  // low 32b = LDS offset
        asm volatile("global_load_async_to_lds_b128 %0, %1, off"
                     :: "v"(la), "v"(ga) : "memory");
    }
    if (tid < TAIL_COUNT) {
        const int idx = tid + TAIL_BASE;
        uint64_t ga = (uint64_t)(uintptr_t)(gsrc + idx);
        uint32_t la = (uint32_t)(uintptr_t)(ldst + idx);
        asm volatile("global_load_async_to_lds_b128 %0, %1, off"
                     :: "v"(la), "v"(ga) : "memory");
    }

    // ---- Overlap the dependent label->x[label] gather with the staging.
    float xlbl = 0.0f;
    if (tid == 0) {
        int lbl = label[0];            // low word correct for int32 or int64
        xlbl = x[lbl];                 // x[0, label[0]]
    }

    // Each wave consumes only the LDS slice it staged itself, so its own
    // ASYNCcnt==0 is sufficient ordering (no workgroup barrier needed here).
    asm volatile("s_wait_asynccnt 0" ::: "memory");

    // ---- exp + per-lane fp32 accumulation (unnormalized, like the reference)
    float s = 0.0f;
    #pragma unroll
    for (int k = 0; k < FULL_ITERS; ++k) {
        float4 v = ldst[tid + k * BLOCK];               // ds_load_b128
        s += __expf(v.x) + __expf(v.y) + __expf(v.z) + __expf(v.w);
    }
    if (tid < TAIL_COUNT) {
        float4 v = ldst[tid + TAIL_BASE];
        s += __expf(v.x) + __expf(v.y) + __expf(v.z) + __expf(v.w);
    }

    // ---- wave32 tree reduction (lane shuffles)
    #pragma unroll
    for (int off = 16; off > 0; off >>= 1)
        s += __shfl_xor(s, off, 32);

    if ((tid & 31) == 0) wsum[tid >> 5] = s;
    __syncthreads();

    // ---- cross-wave reduction in wave 0, then the final scalar
    if (tid < 32) {
        float t = wsum[tid];           // WAVES == 32, all entries valid
        #pragma unroll
        for (int off = 16; off > 0; off >>= 1)
            t += __shfl_xor(t, off, 32);
        if (tid == 0)
            out[0] = __expf(xlbl) / t;
    }
}

extern "C" void kernel_launch(void* const* d_in, const int* in_sizes, int n_in,
                              void* d_out, int out_size, void* d_ws, size_t ws_size,
                              hipStream_t stream)
{
    (void)in_sizes; (void)n_in; (void)out_size; (void)d_ws; (void)ws_size;
    const float* x     = (const float*)d_in[0];
    const int*   label = (const int*)d_in[1];
    float*       out   = (float*)d_out;

    softmax_row0_prob_kernel<<<1, BLOCK, 0, stream>>>(x, label, out);
}